// GDN_OG_27504970563973
// MI455X (gfx1250) — compile-verified
//
#include <hip/hip_runtime.h>
#include <math.h>

// Sizes from the reference
#define BB   64
#define NN   1024
#define FIN  16
#define DIMD 64
#define KK   20
#define BN   (BB * NN)           // 65536
#define EPSV 1e-5f
#define NEG_SLOPE 0.2f
#define SBLK 256                 // stats partial blocks (fixed for determinism)

typedef __attribute__((ext_vector_type(2))) float v2f;
typedef __attribute__((ext_vector_type(8))) float v8f;

// ---------------- K0: normalize embedding + emb-side score terms ----------------
__global__ void k0_norm_emb(const float* __restrict__ emb,
                            const float* __restrict__ att_em_i,
                            const float* __restrict__ att_em_j,
                            float* __restrict__ emb_norm,
                            float* __restrict__ e_si,
                            float* __restrict__ e_sj) {
    __shared__ float ls[DIMD];
    int n = blockIdx.x, d = threadIdx.x;
    float e = emb[n * DIMD + d];

    ls[d] = e * e;
    __syncthreads();
    for (int s = 32; s > 0; s >>= 1) { if (d < s) ls[d] += ls[d + s]; __syncthreads(); }
    float nrm = sqrtf(ls[0]);
    __syncthreads();
    emb_norm[n * DIMD + d] = e / nrm;

    ls[d] = e * att_em_i[d];
    __syncthreads();
    for (int s = 32; s > 0; s >>= 1) { if (d < s) ls[d] += ls[d + s]; __syncthreads(); }
    if (d == 0) e_si[n] = ls[0];
    __syncthreads();

    ls[d] = e * att_em_j[d];
    __syncthreads();
    for (int s = 32; s > 0; s >>= 1) { if (d < s) ls[d] += ls[d + s]; __syncthreads(); }
    if (d == 0) e_sj[n] = ls[0];
}

// ---------------- K1: cos = En @ En^T, WMMA f32 16x16x4, 1 wave / 16x16 tile ----
__global__ void k1_cos_wmma(const float* __restrict__ E, float* __restrict__ cosm) {
    int tid = threadIdx.x;                 // 0..31
    int tm = blockIdx.x >> 6;              // row tile 0..63
    int tn = blockIdx.x & 63;              // col tile 0..63
    int lane16 = tid & 15;
    int khalf  = (tid >> 4) * 2;           // lanes 0-15: K={0,1}; lanes 16-31: K={2,3}

    const float* arow = E + (size_t)(tm * 16 + lane16) * DIMD;  // A: M = lane16
    const float* brow = E + (size_t)(tn * 16 + lane16) * DIMD;  // B = E^T: N = lane16

    v8f c = {};
#pragma unroll
    for (int kc = 0; kc < DIMD / 4; ++kc) {
        v2f a, b;
        a.x = arow[kc * 4 + khalf];
        a.y = arow[kc * 4 + khalf + 1];
        b.x = brow[kc * 4 + khalf];
        b.y = brow[kc * 4 + khalf + 1];
        c = __builtin_amdgcn_wmma_f32_16x16x4_f32(false, a, false, b,
                                                  (short)0, c, false, false);
    }
    int col   = tn * 16 + lane16;
    int rbase = tm * 16 + ((tid >= 16) ? 8 : 0);
#pragma unroll
    for (int r = 0; r < 8; ++r)
        cosm[(size_t)(rbase + r) * NN + col] = c[r];
}

// ---------------- K2: top-20 per row (LDS row + 20 wave-argmax passes) ---------
__global__ void k2_topk(const float* __restrict__ cosm, int* __restrict__ idxb) {
    __shared__ float row[NN];
    int n = blockIdx.x, t = threadIdx.x;   // 32 threads
    for (int j = t; j < NN; j += 32) row[j] = cosm[(size_t)n * NN + j];
    __syncthreads();
    for (int it = 0; it < KK; ++it) {
        float v = -1e30f; int vi = 0;
        for (int j = t; j < NN; j += 32) {
            float x = row[j];
            if (x > v) { v = x; vi = j; }  // ascending j -> lowest index on tie
        }
        for (int off = 16; off > 0; off >>= 1) {
            float ov = __shfl_down(v, off, 32);
            int   oi = __shfl_down(vi, off, 32);
            if (ov > v || (ov == v && oi < vi)) { v = ov; vi = oi; }
        }
        vi = __shfl(vi, 0, 32);
        if (t == 0) { idxb[n * KK + it] = vi; row[vi] = -1e30f; }
        __syncthreads();
    }
}

// ---------------- K3: h = data @ lin_w^T, WMMA f32 16x16x4 ---------------------
// 1 wave per 16-row (b,n) tile; 4 column tiles cover DIM=64; K loop = FIN/4 = 4.
__global__ void k3_h_wmma(const float* __restrict__ data,
                          const float* __restrict__ lin_w,
                          float* __restrict__ h) {
    int tid = threadIdx.x;
    int t   = blockIdx.x;                  // bn tile 0..4095
    int lane16 = tid & 15;
    int khalf  = (tid >> 4) * 2;

    const float* arow = data + (size_t)(t * 16 + lane16) * FIN;
    v8f acc[4] = {};
#pragma unroll
    for (int kc = 0; kc < FIN / 4; ++kc) {
        v2f a;
        a.x = arow[kc * 4 + khalf];
        a.y = arow[kc * 4 + khalf + 1];
#pragma unroll
        for (int ct = 0; ct < 4; ++ct) {
            int dcol = ct * 16 + lane16;
            v2f b;                         // B[f][d] = lin_w[d*FIN + f]
            b.x = lin_w[dcol * FIN + kc * 4 + khalf];
            b.y = lin_w[dcol * FIN + kc * 4 + khalf + 1];
            acc[ct] = __builtin_amdgcn_wmma_f32_16x16x4_f32(false, a, false, b,
                                                            (short)0, acc[ct], false, false);
        }
    }
    int rbase = t * 16 + ((tid >= 16) ? 8 : 0);
#pragma unroll
    for (int ct = 0; ct < 4; ++ct)
#pragma unroll
        for (int r = 0; r < 8; ++r)
            h[(size_t)(rbase + r) * DIMD + ct * 16 + lane16] = acc[ct][r];
}

// ---------------- K4: s_i, s_j per (b,n) ---------------------------------------
__global__ void k4_scores(const float* __restrict__ h,
                          const float* __restrict__ att_i,
                          const float* __restrict__ att_j,
                          const float* __restrict__ e_si,
                          const float* __restrict__ e_sj,
                          float* __restrict__ si, float* __restrict__ sj) {
    for (int bn = blockIdx.x * blockDim.x + threadIdx.x; bn < BN;
         bn += gridDim.x * blockDim.x) {
        int n = bn & (NN - 1);
        float ai = e_si[n], aj = e_sj[n];
        const float* hr = h + (size_t)bn * DIMD;
#pragma unroll
        for (int d = 0; d < DIMD; ++d) {
            float x = hr[d];
            ai += x * att_i[d];
            aj += x * att_j[d];
        }
        si[bn] = ai; sj[bn] = aj;
    }
}

// ---------------- K5: leaky-relu + softmax + neighbor aggregation --------------
__global__ void k5_attn(const float* __restrict__ h,
                        const int* __restrict__ idxb,
                        const float* __restrict__ si,
                        const float* __restrict__ sj,
                        const float* __restrict__ gnn_bias,
                        float* __restrict__ out1) {
    __shared__ float s_alpha[KK];
    __shared__ int   s_nb[KK];
    int bn = blockIdx.x, t = threadIdx.x;  // 64 threads = one channel each
    int b = bn >> 10, n = bn & (NN - 1);
    if (t < KK) {
        int nb = idxb[n * KK + t];
        s_nb[t] = nb;
        float a = si[bn] + sj[b * NN + nb];
        s_alpha[t] = (a > 0.f) ? a : NEG_SLOPE * a;
    }
    __syncthreads();
    if (t == 0) {
        float m = s_alpha[0];
        for (int k = 1; k < KK; ++k) m = fmaxf(m, s_alpha[k]);
        float s = 0.f;
        for (int k = 0; k < KK; ++k) { float e = __expf(s_alpha[k] - m); s_alpha[k] = e; s += e; }
        float inv = 1.f / s;
        for (int k = 0; k < KK; ++k) s_alpha[k] *= inv;
    }
    __syncthreads();
    float acc = 0.f;
#pragma unroll
    for (int k = 0; k < KK; ++k)
        acc += s_alpha[k] * h[(size_t)(b * NN + s_nb[k]) * DIMD + t];
    out1[(size_t)bn * DIMD + t] = acc + gnn_bias[t];
}

// ---------------- deterministic per-channel sum/sumsq reduction ----------------
// Stride (gridDim*256) is a multiple of 64 -> each thread's channel is fixed (t&63).
__global__ void k_stats_partial(const float* __restrict__ x, float* __restrict__ partial) {
    __shared__ float ls[512];
    int t = threadIdx.x;                   // 256
    float s = 0.f, sq = 0.f;
    for (size_t i = (size_t)blockIdx.x * 256 + t; i < (size_t)BN * DIMD;
         i += (size_t)gridDim.x * 256) {
        float v = x[i]; s += v; sq += v * v;
    }
    ls[t] = s; ls[256 + t] = sq;
    __syncthreads();
    if (t < DIMD) {
        float ss = ls[t] + ls[t + 64] + ls[t + 128] + ls[t + 192];
        float qq = ls[256 + t] + ls[256 + t + 64] + ls[256 + t + 128] + ls[256 + t + 192];
        partial[blockIdx.x * 2 * DIMD + t]        = ss;
        partial[blockIdx.x * 2 * DIMD + DIMD + t] = qq;
    }
}

__global__ void k_stats_final(const float* __restrict__ partial, float* __restrict__ stats) {
    int t = threadIdx.x;                   // 128 = 2*DIM
    float s = 0.f;
    for (int b = 0; b < SBLK; ++b) s += partial[b * 2 * DIMD + t];
    stats[t] = s;
}

// ---------------- K7: bn1 + relu, multiply by emb -> rep -----------------------
__global__ void k7_bn1(const float* __restrict__ out1,
                       const float* __restrict__ stats1,
                       const float* __restrict__ emb,
                       const float* __restrict__ g1, const float* __restrict__ b1,
                       float* __restrict__ rep) {
    const float invc = 1.f / (float)BN;
    for (size_t i = (size_t)blockIdx.x * blockDim.x + threadIdx.x; i < (size_t)BN * DIMD;
         i += (size_t)gridDim.x * blockDim.x) {
        int d = (int)(i & (DIMD - 1));
        int n = (int)((i >> 6) & (NN - 1));
        float mean = stats1[d] * invc;
        float var  = stats1[DIMD + d] * invc - mean * mean;
        float rs   = rsqrtf(var + EPSV);
        float y = (out1[i] - mean) * rs * g1[d] + b1[d];
        y = fmaxf(y, 0.f);
        rep[i] = y * emb[n * DIMD + d];
    }
}

// ---------------- K9: bn2 + relu + final projection ----------------------------
__global__ void k9_out(const float* __restrict__ rep,
                       const float* __restrict__ stats2,
                       const float* __restrict__ g2, const float* __restrict__ b2,
                       const float* __restrict__ out_w, const float* __restrict__ out_b,
                       float* __restrict__ out) {
    __shared__ float ls[DIMD];
    int bn = blockIdx.x, d = threadIdx.x;  // 64 threads
    const float invc = 1.f / (float)BN;
    float mean = stats2[d] * invc;
    float var  = stats2[DIMD + d] * invc - mean * mean;
    float rs   = rsqrtf(var + EPSV);
    float z = (rep[(size_t)bn * DIMD + d] - mean) * rs * g2[d] + b2[d];
    z = fmaxf(z, 0.f);
    ls[d] = z * out_w[d];
    __syncthreads();
    for (int s = 32; s > 0; s >>= 1) { if (d < s) ls[d] += ls[d + s]; __syncthreads(); }
    if (d == 0) out[bn] = ls[0] + out_b[0];
}

// ---------------- workspace layout (float units) -------------------------------
static const size_t OFF_COS  = 0;                          // 1024*1024
static const size_t OFF_EN   = OFF_COS  + (size_t)NN * NN; // 1024*64
static const size_t OFF_ESI  = OFF_EN   + (size_t)NN * DIMD;
static const size_t OFF_ESJ  = OFF_ESI  + NN;
static const size_t OFF_SI   = OFF_ESJ  + NN;
static const size_t OFF_SJ   = OFF_SI   + BN;
static const size_t OFF_H    = OFF_SJ   + BN;              // 16 MB, reused as rep
static const size_t OFF_OUT1 = OFF_H    + (size_t)BN * DIMD;
static const size_t OFF_P1   = OFF_OUT1 + (size_t)BN * DIMD;
static const size_t OFF_P2   = OFF_P1   + (size_t)SBLK * 2 * DIMD;
static const size_t OFF_S1   = OFF_P2   + (size_t)SBLK * 2 * DIMD;
static const size_t OFF_S2   = OFF_S1   + 2 * DIMD;
static const size_t OFF_IDX  = OFF_S2   + 2 * DIMD;        // ints from here (N*K)

extern "C" void kernel_launch(void* const* d_in, const int* in_sizes, int n_in,
                              void* d_out, int out_size, void* d_ws, size_t ws_size,
                              hipStream_t stream) {
    (void)in_sizes; (void)n_in; (void)out_size; (void)ws_size;
    const float* data     = (const float*)d_in[0];
    // d_in[1] = org_edge_index: unused by the reference
    const float* embedding= (const float*)d_in[2];
    const float* lin_w    = (const float*)d_in[3];
    const float* att_i    = (const float*)d_in[4];
    const float* att_j    = (const float*)d_in[5];
    const float* att_em_i = (const float*)d_in[6];
    const float* att_em_j = (const float*)d_in[7];
    const float* gnn_bias = (const float*)d_in[8];
    const float* bn1_g    = (const float*)d_in[9];
    const float* bn1_b    = (const float*)d_in[10];
    const float* bn2_g    = (const float*)d_in[11];
    const float* bn2_b    = (const float*)d_in[12];
    const float* out_w    = (const float*)d_in[13];
    const float* out_b    = (const float*)d_in[14];
    float* out = (float*)d_out;

    float* ws = (float*)d_ws;
    float* cosm = ws + OFF_COS;
    float* En   = ws + OFF_EN;
    float* e_si = ws + OFF_ESI;
    float* e_sj = ws + OFF_ESJ;
    float* si   = ws + OFF_SI;
    float* sj   = ws + OFF_SJ;
    float* h    = ws + OFF_H;
    float* rep  = ws + OFF_H;       // reuse: h dead after K5
    float* out1 = ws + OFF_OUT1;
    float* p1   = ws + OFF_P1;
    float* p2   = ws + OFF_P2;
    float* s1   = ws + OFF_S1;
    float* s2   = ws + OFF_S2;
    int*   idxb = (int*)(ws + OFF_IDX);

    k0_norm_emb<<<NN, DIMD, 0, stream>>>(embedding, att_em_i, att_em_j, En, e_si, e_sj);
    k1_cos_wmma<<<(NN / 16) * (NN / 16), 32, 0, stream>>>(En, cosm);
    k2_topk<<<NN, 32, 0, stream>>>(cosm, idxb);
    k3_h_wmma<<<BN / 16, 32, 0, stream>>>(data, lin_w, h);
    k4_scores<<<64, 256, 0, stream>>>(h, att_i, att_j, e_si, e_sj, si, sj);
    k5_attn<<<BN, DIMD, 0, stream>>>(h, idxb, si, sj, gnn_bias, out1);
    k_stats_partial<<<SBLK, 256, 0, stream>>>(out1, p1);
    k_stats_final<<<1, 2 * DIMD, 0, stream>>>(p1, s1);
    k7_bn1<<<2048, 256, 0, stream>>>(out1, s1, embedding, bn1_g, bn1_b, rep);
    k_stats_partial<<<SBLK, 256, 0, stream>>>(rep, p2);
    k_stats_final<<<1, 2 * DIMD, 0, stream>>>(p2, s2);
    k9_out<<<BN, DIMD, 0, stream>>>(rep, s2, bn2_g, bn2_b, out_w, out_b, out);
}